// ESAGEConv_26070451487319
// MI455X (gfx1250) — compile-verified
//
#include <hip/hip_runtime.h>
#include <math.h>

// ---- problem constants (from reference) ----
#define N_NODES 50000
#define N_EDGES 800000
#define NIN  64
#define EIN  32
#define NOUT 64
#define EOUT 32

typedef __attribute__((ext_vector_type(2))) float v2f;
typedef __attribute__((ext_vector_type(8))) float v8f;

// f32 WMMA: D(16x16) = A(16x4) * B(4x16) + C, full fp32 (matches reference math)
static __device__ __forceinline__ v8f wmma_f32(v2f a, v2f b, v8f c) {
  return __builtin_amdgcn_wmma_f32_16x16x4_f32(false, a, false, b, (short)0, c, false, false);
}

// ---------------- zero the accumulator regions (sums + cnt) ----------------
__global__ void zero_kernel(float* __restrict__ p, int n) {
  int i = blockIdx.x * blockDim.x + threadIdx.x;
  int stride = gridDim.x * blockDim.x;
  for (; i < n; i += stride) p[i] = 0.0f;
}

// ---------------- pass 1: message (fused gather+GEMM+tanh) + scatter-add ----
// wave tile: M=16 edges, N=64 outputs, K=96 (feat[src] 64 | edge_feat 32)
__global__ __launch_bounds__(256) void msg_scatter_kernel(
    const float* __restrict__ feat, const float* __restrict__ edge_feat,
    const float* __restrict__ w_msg, const float* __restrict__ b_msg,
    const int* __restrict__ src_idx, const int* __restrict__ dst_idx,
    float* sums, float* cnt)
{
  // w_msg (96x64) staged as K-pairs: lds_w[p*64+n] = {w[2p][n], w[2p+1][n]}
  __shared__ v2f lds_w[48 * 64];   // 24 KB
  int tid = threadIdx.x;
  for (int i = tid; i < 48 * 64; i += 256) {
    int p = i >> 6, n = i & 63;
    v2f v; v.x = w_msg[(2 * p) * NOUT + n]; v.y = w_msg[(2 * p + 1) * NOUT + n];
    lds_w[i] = v;
  }
  __syncthreads();

  int wave = tid >> 5, lane = tid & 31;
  int half = lane >> 4, mrow = lane & 15;
  int edgeBase = (blockIdx.x * 8 + wave) * 16;

  int myEdge = edgeBase + mrow;
  const float* rowS = feat + (size_t)src_idx[myEdge] * NIN;
  const float* rowE = edge_feat + (size_t)myEdge * EIN;

  v8f acc[4] = {};
#pragma unroll
  for (int k = 0; k < 96; k += 4) {
    int c = k + 2 * half;                 // both lane-halves land on same side of 64
    const float* pa = (c < 64) ? (rowS + c) : (rowE + (c - 64));
    v2f a = *(const v2f*)pa;
    int pb = (c >> 1) * 64 + mrow;
#pragma unroll
    for (int t = 0; t < 4; ++t)
      acc[t] = wmma_f32(a, lds_w[pb + t * 16], acc[t]);
  }

  int dstv[8];
#pragma unroll
  for (int r = 0; r < 8; ++r) dstv[r] = dst_idx[edgeBase + r + 8 * half];

#pragma unroll
  for (int t = 0; t < 4; ++t) {
    int n = t * 16 + mrow;
    float bias = b_msg[n];
#pragma unroll
    for (int r = 0; r < 8; ++r) {
      float v = tanhf(acc[t][r] + bias);
      atomicAdd(&sums[(size_t)dstv[r] * NOUT + n], v);
    }
  }
  if (lane < 16) atomicAdd(&cnt[dst_idx[edgeBase + lane]], 1.0f);
}

// ---------------- pass 2: node update (in-place sums->h) -------------------
// wave tile: M=16 nodes, N=64, K=128 (neigh 64 | feat 64)
// hs aliases: reads sums rows, then overwrites same rows with h (per-wave safe)
__global__ __launch_bounds__(256) void node_update_kernel(
    const float* __restrict__ feat, const float* __restrict__ w_node,
    const float* __restrict__ b_node, float* hs, const float* cnt)
{
  __shared__ v2f lds_w[64 * 64];   // 32 KB
  int tid = threadIdx.x;
  for (int i = tid; i < 64 * 64; i += 256) {
    int p = i >> 6, n = i & 63;
    v2f v; v.x = w_node[(2 * p) * NOUT + n]; v.y = w_node[(2 * p + 1) * NOUT + n];
    lds_w[i] = v;
  }
  __syncthreads();

  int wave = tid >> 5, lane = tid & 31;
  int tile = blockIdx.x * 8 + wave;
  if (tile >= N_NODES / 16) return;
  int half = lane >> 4, mrow = lane & 15;
  int nodeBase = tile * 16;
  int node = nodeBase + mrow;

  float invc = 1.0f / fmaxf(cnt[node], 1.0f);
  const float* rowSum = hs + (size_t)node * NOUT;   // sums (pre-overwrite)
  const float* rowF = feat + (size_t)node * NIN;

  v8f acc[4] = {};
#pragma unroll
  for (int k = 0; k < 128; k += 4) {
    int c = k + 2 * half;
    bool isN = (c < 64);
    const float* pa = isN ? (rowSum + c) : (rowF + (c - 64));
    float sc = isN ? invc : 1.0f;
    v2f a = *(const v2f*)pa;
    a.x *= sc; a.y *= sc;
    int pb = (c >> 1) * 64 + mrow;
#pragma unroll
    for (int t = 0; t < 4; ++t)
      acc[t] = wmma_f32(a, lds_w[pb + t * 16], acc[t]);
  }

#pragma unroll
  for (int t = 0; t < 4; ++t) {
    int n = t * 16 + mrow;
    float bias = b_node[n];
#pragma unroll
    for (int r = 0; r < 8; ++r)
      hs[(size_t)(nodeBase + r + 8 * half) * NOUT + n] = tanhf(acc[t][r] + bias);
  }
}

// ---------------- pass 3: edge update --------------------------------------
// wave tile: M=16 edges, N=32, K=160 (feat[src] 64 | h[dst] 64 | edge_feat 32)
__global__ __launch_bounds__(256) void edge_update_kernel(
    const float* __restrict__ feat, const float* __restrict__ edge_feat,
    const float* __restrict__ w_edge, const float* __restrict__ b_edge,
    const int* __restrict__ src_idx, const int* __restrict__ dst_idx,
    const float* __restrict__ h, float* __restrict__ e)
{
  __shared__ v2f lds_w[80 * 32];   // 20 KB: K-pairs of w_edge (160x32)
  int tid = threadIdx.x;
  for (int i = tid; i < 80 * 32; i += 256) {
    int p = i >> 5, n = i & 31;
    v2f v; v.x = w_edge[(2 * p) * EOUT + n]; v.y = w_edge[(2 * p + 1) * EOUT + n];
    lds_w[i] = v;
  }
  __syncthreads();

  int wave = tid >> 5, lane = tid & 31;
  int half = lane >> 4, mrow = lane & 15;
  int edgeBase = (blockIdx.x * 8 + wave) * 16;
  int myEdge = edgeBase + mrow;

  const float* rowS = feat + (size_t)src_idx[myEdge] * NIN;
  const float* rowH = h + (size_t)dst_idx[myEdge] * NOUT;
  const float* rowE = edge_feat + (size_t)myEdge * EIN;

  v8f acc[2] = {};
#pragma unroll
  for (int k = 0; k < 160; k += 4) {
    int c = k + 2 * half;
    const float* pa = (c < 64) ? (rowS + c)
                    : (c < 128) ? (rowH + (c - 64))
                                : (rowE + (c - 128));
    v2f a = *(const v2f*)pa;
    int pb = (c >> 1) * 32 + mrow;
    acc[0] = wmma_f32(a, lds_w[pb], acc[0]);
    acc[1] = wmma_f32(a, lds_w[pb + 16], acc[1]);
  }

#pragma unroll
  for (int t = 0; t < 2; ++t) {
    int n = t * 16 + mrow;
    float bias = b_edge[n];
#pragma unroll
    for (int r = 0; r < 8; ++r)
      e[(size_t)(edgeBase + r + 8 * half) * EOUT + n] = tanhf(acc[t][r] + bias);
  }
}

extern "C" void kernel_launch(void* const* d_in, const int* in_sizes, int n_in,
                              void* d_out, int out_size, void* d_ws, size_t ws_size,
                              hipStream_t stream) {
  const float* feat      = (const float*)d_in[0];
  const float* edge_feat = (const float*)d_in[1];
  const float* w_msg     = (const float*)d_in[2];
  const float* b_msg     = (const float*)d_in[3];
  const float* w_node    = (const float*)d_in[4];
  const float* b_node    = (const float*)d_in[5];
  const float* w_edge    = (const float*)d_in[6];
  const float* b_edge    = (const float*)d_in[7];
  const int* src_idx     = (const int*)d_in[8];
  const int* dst_idx     = (const int*)d_in[9];

  // Output layout: h (50000*64) then e (800000*32).
  // Accumulators alias d_out (no d_ws dependence):
  //   sums -> h region (node pass converts sums->h in place),
  //   cnt  -> first 50000 floats of e region (e fully rewritten in pass 3).
  float* h = (float*)d_out;
  float* e = h + (size_t)N_NODES * NOUT;
  float* sums = h;
  float* cnt  = e;

  // zero sums+cnt (contiguous) every launch -> deterministic under replay
  int zn = N_NODES * NOUT + N_NODES;
  hipLaunchKernelGGL(zero_kernel, dim3(512), dim3(256), 0, stream, sums, zn);

  hipLaunchKernelGGL(msg_scatter_kernel, dim3(N_EDGES / 128), dim3(256), 0, stream,
                     feat, edge_feat, w_msg, b_msg, src_idx, dst_idx, sums, cnt);

  hipLaunchKernelGGL(node_update_kernel, dim3((N_NODES / 16 + 7) / 8), dim3(256), 0, stream,
                     feat, w_node, b_node, h, cnt);

  hipLaunchKernelGGL(edge_update_kernel, dim3(N_EDGES / 128), dim3(256), 0, stream,
                     feat, edge_feat, w_edge, b_edge, src_idx, dst_idx, h, e);
}